// MultiHeadGraphAttention_42683384988232
// MI455X (gfx1250) — compile-verified
//
#include <hip/hip_runtime.h>
#include <hip/hip_bf16.h>

#define GN      6144
#define GHID    48
#define GHEADS  3
#define GHEAD   16

typedef __attribute__((ext_vector_type(2))) float v2f;
typedef __attribute__((ext_vector_type(8))) float v8f;

__device__ __forceinline__ v8f wmma_f32_k4(v2f a, v2f b, v8f c) {
    // D(16x16,f32) = A(16x4,f32) x B(4x16,f32) + C
    return __builtin_amdgcn_wmma_f32_16x16x4_f32(
        /*neg_a=*/false, a, /*neg_b=*/false, b,
        /*c_mod=*/(short)0, c, /*reuse_a=*/false, /*reuse_b=*/false);
}

// ---------------------------------------------------------------------------
// Kernel 1: QKV projection. q = h @ Wq^T + bq (torch Linear), stored head-major
// [HEADS][N][HEAD] so the attention kernel reads contiguous 16-float rows.
// ---------------------------------------------------------------------------
__global__ __launch_bounds__(256) void qkv_proj_kernel(
    const float* __restrict__ h,
    const float* __restrict__ Wq, const float* __restrict__ bq,
    const float* __restrict__ Wk, const float* __restrict__ bk,
    const float* __restrict__ Wv, const float* __restrict__ bv,
    float* __restrict__ Q, float* __restrict__ K, float* __restrict__ V)
{
    int idx = blockIdx.x * blockDim.x + threadIdx.x;
    if (idx >= GN * GHID) return;
    int n = idx / GHID;
    int j = idx - n * GHID;

    const float* hr = h + n * GHID;
    const float* wq = Wq + j * GHID;
    const float* wk = Wk + j * GHID;
    const float* wv = Wv + j * GHID;

    float aq = bq[j], ak = bk[j], av = bv[j];
#pragma unroll
    for (int i = 0; i < GHID; ++i) {
        float hv = hr[i];
        aq = fmaf(hv, wq[i], aq);
        ak = fmaf(hv, wk[i], ak);
        av = fmaf(hv, wv[i], av);
    }
    int head = j / GHEAD;
    int d    = j - head * GHEAD;
    int o    = (head * GN + n) * GHEAD + d;
    Q[o] = aq; K[o] = ak; V[o] = av;
}

// ---------------------------------------------------------------------------
// Kernel 2: masked attention + dense softmax scores, fp32 WMMA.
// Block = 128 threads (4 waves). Block owns 16 query rows of one head.
// Wave w owns column tiles mt = w, w+4, ... (96 tiles of 16 columns each).
// ---------------------------------------------------------------------------
__global__ __launch_bounds__(128) void attn_kernel(
    const float* __restrict__ A,
    const float* __restrict__ Q,
    const float* __restrict__ K,
    const float* __restrict__ V,
    float* __restrict__ out,
    float* __restrict__ scores)
{
    const int row0 = blockIdx.x * 16;
    const int head = blockIdx.y;
    const int tid  = threadIdx.x;
    const int wave = tid >> 5;
    const int lane = tid & 31;
    const int l16  = lane & 15;
    const int hi   = lane >> 4;     // 0: rows r, 1: rows r+8 (C/D layout)
    const int off  = hi * 2;        // A/B operand K-offset for upper half

    const float scale = sqrtf((float)GN);

    const float* Qh = Q + (size_t)head * GN * GHEAD;
    const float* Kh = K + (size_t)head * GN * GHEAD;
    const float* Vh = V + (size_t)head * GN * GHEAD;
    float*       sc = scores + (size_t)head * GN * GN;

    __shared__ float red[4][16][2];   // per-wave (max, sum) per tile row
    __shared__ float rowstat[16][2];  // final (max, 1/sum) per tile row
    __shared__ float ptile[4][256];   // per-wave P tile for layout transpose
    __shared__ float obuf[4][256];    // per-wave O accumulators

    // A-operand for S = Q*K^T: lane holds Q[row0+l16][c*4+off .. +1]
    v2f qa[4];
    {
        const float* qrow = Qh + (size_t)(row0 + l16) * GHEAD;
#pragma unroll
        for (int c = 0; c < 4; ++c)
            qa[c] = *(const v2f*)(qrow + c * 4 + off);
    }

    // ---------------- Pass 1: online (max, sum-of-exp) over all columns -----
    float m8[8], s8[8];
#pragma unroll
    for (int r = 0; r < 8; ++r) { m8[r] = -__builtin_inff(); s8[r] = 0.0f; }

    for (int mt = wave; mt < GN / 16; mt += 4) {
        const int col0 = mt * 16;
        const float* krow = Kh + (size_t)(col0 + l16) * GHEAD;
        v2f kb[4];
#pragma unroll
        for (int c = 0; c < 4; ++c)
            kb[c] = *(const v2f*)(krow + c * 4 + off);

        v8f acc = {};
#pragma unroll
        for (int c = 0; c < 4; ++c)
            acc = wmma_f32_k4(qa[c], kb[c], acc);

        const float* arow = A + (size_t)(row0 + 8 * hi) * GN + col0 + l16;
#pragma unroll
        for (int r = 0; r < 8; ++r) {
            float av = arow[(size_t)r * GN];
            float s  = acc[r] * av * scale;       // masked, scaled score
            float nm = fmaxf(m8[r], s);
            s8[r] = s8[r] * __expf(m8[r] - nm) + __expf(s - nm);
            m8[r] = nm;
        }
    }

    // Reduce (max,sum) across the 16 lanes of each half-wave.
#pragma unroll
    for (int r = 0; r < 8; ++r) {
#pragma unroll
        for (int d = 1; d <= 8; d <<= 1) {
            float om = __shfl_xor(m8[r], d, 32);
            float os = __shfl_xor(s8[r], d, 32);
            float nm = fmaxf(m8[r], om);
            s8[r] = s8[r] * __expf(m8[r] - nm) + os * __expf(om - nm);
            m8[r] = nm;
        }
    }
    if (l16 == 0) {                 // lane 0 -> rows 0..7, lane 16 -> rows 8..15
#pragma unroll
        for (int r = 0; r < 8; ++r) {
            red[wave][8 * hi + r][0] = m8[r];
            red[wave][8 * hi + r][1] = s8[r];
        }
    }
    __syncthreads();

    if (tid < 16) {                 // combine the 4 waves per tile row
        float M = red[0][tid][0], S = red[0][tid][1];
#pragma unroll
        for (int w = 1; w < 4; ++w) {
            float om = red[w][tid][0], os = red[w][tid][1];
            float nm = fmaxf(M, om);
            S = S * __expf(M - nm) + os * __expf(om - nm);
            M = nm;
        }
        rowstat[tid][0] = M;
        rowstat[tid][1] = 1.0f / S;
    }
    __syncthreads();

    float rm[8], rv[8];
#pragma unroll
    for (int r = 0; r < 8; ++r) {
        rm[r] = rowstat[8 * hi + r][0];
        rv[r] = rowstat[8 * hi + r][1];
    }

    // ---------------- Pass 2: write normalized scores, accumulate O = P*V ---
    v8f oacc = {};
    float* pt = ptile[wave];

    for (int mt = wave; mt < GN / 16; mt += 4) {
        const int col0 = mt * 16;
        const float* krow = Kh + (size_t)(col0 + l16) * GHEAD;
        v2f kb[4];
#pragma unroll
        for (int c = 0; c < 4; ++c)
            kb[c] = *(const v2f*)(krow + c * 4 + off);

        v8f acc = {};
#pragma unroll
        for (int c = 0; c < 4; ++c)
            acc = wmma_f32_k4(qa[c], kb[c], acc);

        const float* arow = A  + (size_t)(row0 + 8 * hi) * GN + col0 + l16;
        float*       srow = sc + (size_t)(row0 + 8 * hi) * GN + col0 + l16;
#pragma unroll
        for (int r = 0; r < 8; ++r) {
            float av = arow[(size_t)r * GN];
            float s  = acc[r] * av * scale;
            float p  = __expf(s - rm[r]) * rv[r];          // normalized prob
            __builtin_nontemporal_store(p, srow + (size_t)r * GN);
            pt[(8 * hi + r) * 16 + l16] = p;               // D-layout -> LDS
        }

        // O-tile WMMA: A-operand = P (rows=queries) from LDS, B-operand = V.
        const float* vbase = Vh + (size_t)col0 * GHEAD;
        v2f pa[4], vb[4];
#pragma unroll
        for (int c = 0; c < 4; ++c) {
            pa[c].x = pt[l16 * 16 + c * 4 + off];
            pa[c].y = pt[l16 * 16 + c * 4 + off + 1];
            vb[c].x = vbase[(c * 4 + off)     * GHEAD + l16];
            vb[c].y = vbase[(c * 4 + off + 1) * GHEAD + l16];
        }
#pragma unroll
        for (int c = 0; c < 4; ++c)
            oacc = wmma_f32_k4(pa[c], vb[c], oacc);
    }

    // Combine partial O across waves, write out[N][48].
    float* ob = obuf[wave];
#pragma unroll
    for (int r = 0; r < 8; ++r)
        ob[(8 * hi + r) * 16 + l16] = oacc[r];
    __syncthreads();

    for (int i = tid; i < 256; i += 128) {
        float v = obuf[0][i] + obuf[1][i] + obuf[2][i] + obuf[3][i];
        int row = i >> 4, d = i & 15;
        out[(size_t)(row0 + row) * GHID + head * GHEAD + d] = v;
    }
}

// ---------------------------------------------------------------------------
extern "C" void kernel_launch(void* const* d_in, const int* in_sizes, int n_in,
                              void* d_out, int out_size, void* d_ws, size_t ws_size,
                              hipStream_t stream)
{
    const float* A  = (const float*)d_in[0];
    const float* h  = (const float*)d_in[1];
    const float* Wq = (const float*)d_in[2];
    const float* bq = (const float*)d_in[3];
    const float* Wk = (const float*)d_in[4];
    const float* bk = (const float*)d_in[5];
    const float* Wv = (const float*)d_in[6];
    const float* bv = (const float*)d_in[7];

    float* out    = (float*)d_out;                       // [N, 48]
    float* scores = out + (size_t)GN * GHID;             // [H, N, N]

    const size_t per = (size_t)GHEADS * GN * GHEAD;      // 294912 floats
    float* Q = (float*)d_ws;
    float* K = Q + per;
    float* V = K + per;

    qkv_proj_kernel<<<(GN * GHID + 255) / 256, 256, 0, stream>>>(
        h, Wq, bq, Wk, bk, Wv, bv, Q, K, V);

    dim3 grid(GN / 16, GHEADS);
    attn_kernel<<<grid, 128, 0, stream>>>(A, Q, K, V, out, scores);
}